// GraphCapsuleConv_87024627352057
// MI455X (gfx1250) — compile-verified
//
#include <hip/hip_runtime.h>
#include <cstdint>
#include <cstddef>

typedef __attribute__((ext_vector_type(2))) float v2f;
typedef __attribute__((ext_vector_type(8))) float v8f;

#define D 128   // D_IN == D_HID == 128

__device__ __forceinline__ float selu_f(float x) {
    // selu(x) = scale * (x>0 ? x : alpha*(exp(x)-1))
    // scale = 1.0507009873554805, alpha = 1.6732632423543772, scale*alpha below
    return x > 0.0f ? 1.0507009873554805f * x
                    : 1.7580993408473766f * (__expf(x) - 1.0f);
}

// -------- Kernel 1: CSR row offsets from sorted COO rows (binary search) ----
__global__ __launch_bounds__(256) void rowptr_kernel(
    const int* __restrict__ erow, int* __restrict__ row_ptr, int N, int E) {
    const int r = blockIdx.x * 256 + threadIdx.x;
    if (r > N) return;
    int lo = 0, hi = E;
    while (lo < hi) {
        const int mid = (lo + hi) >> 1;
        if (erow[mid] < r) lo = mid + 1; else hi = mid;
    }
    row_ptr[r] = lo;
}

// -------- Kernel 2: SPMM + residual for BOTH branches (x and x^2) ----------
// h0[0][row][:] = sum_e val*x[col] + x[row]
// h0[1][row][:] = sum_e val*x[col]^2 + x[row]^2
__global__ __launch_bounds__(256) void spmm_kernel(
    const float* __restrict__ x, const int* __restrict__ row_ptr,
    const int* __restrict__ ecol, const float* __restrict__ eval,
    float* __restrict__ h0, int N) {
    const int lane = threadIdx.x & 31;
    const int row  = blockIdx.x * 8 + (threadIdx.x >> 5);
    if (row >= N) return;
    const int q = lane * 4;                 // 4 features per lane
    const float4 xr = *(const float4*)(x + (size_t)row * D + q);
    float s0x = xr.x, s0y = xr.y, s0z = xr.z, s0w = xr.w;
    float s1x = xr.x * xr.x, s1y = xr.y * xr.y, s1z = xr.z * xr.z, s1w = xr.w * xr.w;
    const int e1 = row_ptr[row + 1];
    for (int e = row_ptr[row]; e < e1; ++e) {
        const float v = eval[e];
        const int   c = ecol[e];
        const float4 xc = *(const float4*)(x + (size_t)c * D + q);
        s0x += v * xc.x;          s0y += v * xc.y;
        s0z += v * xc.z;          s0w += v * xc.w;
        s1x += v * xc.x * xc.x;   s1y += v * xc.y * xc.y;
        s1z += v * xc.z * xc.z;   s1w += v * xc.w * xc.w;
    }
    *(float4*)(h0 + (size_t)row * D + q) = make_float4(s0x, s0y, s0z, s0w);
    *(float4*)(h0 + (size_t)N * D + (size_t)row * D + q) = make_float4(s1x, s1y, s1z, s1w);
}

// -------- Kernel 3: out = selu(A @ W + b), fp32 WMMA 16x16x4 ---------------
// grid.x = row strips of 128, grid.y = branch; block = 256 threads (8 waves),
// one wave computes a 16x128 output strip with 8 independent 16x16 tiles.
//
// W is staged in LDS *pair-interleaved*: element (k, n) at
//   ldsW[(k>>1)*256 + 2*n + (k&1)]
// so a B fragment (two consecutive K values of one column) is a contiguous
// float2 -> single ds_load_b64 straight into the adjacent VGPR pair WMMA
// requires (no v_mov marshalling).
__global__ __launch_bounds__(256) void gemm_selu_kernel(
    const float* __restrict__ Abase, size_t a_br_stride,
    const float* __restrict__ Wbase,
    const float* __restrict__ biasbase,
    float* __restrict__ outbase, int out_row_stride, size_t out_br_off,
    int N) {
    __shared__ __align__(16) float ldsW[D * D];   // 64 KB

    const int tid  = threadIdx.x;
    const int lane = tid & 31;
    const int wv   = tid >> 5;                    // wave id 0..7
    const int br   = blockIdx.y;
    const int row0 = blockIdx.x * 128;

    const float* A    = Abase + (size_t)br * a_br_stride;
    const float* W    = Wbase + (size_t)br * (D * D);
    const float* bias = biasbase + br * D;
    float*       out  = outbase + (size_t)br * out_br_off;

    // cooperative load of W into LDS, pair-interleaved layout
    for (int idx = tid; idx < 64 * 32; idx += 256) {
        const int p = idx >> 5;                   // k-pair 0..63
        const int q = (idx & 31) << 2;            // n base (multiple of 4)
        const float4 g0 = *(const float4*)(W + (size_t)(2 * p)     * D + q);
        const float4 g1 = *(const float4*)(W + (size_t)(2 * p + 1) * D + q);
        float* base = &ldsW[p * 256 + q * 2];
        *(float2*)(base + 0) = make_float2(g0.x, g1.x);
        *(float2*)(base + 2) = make_float2(g0.y, g1.y);
        *(float2*)(base + 4) = make_float2(g0.z, g1.z);
        *(float2*)(base + 6) = make_float2(g0.w, g1.w);
    }
    __syncthreads();

    const int m  = lane & 15;        // row within A tile / col within B tile
    const int hi = lane >> 4;        // half-wave select (A: K+0/1 vs K+2/3)

    // A fragments for this wave's 16-row strip (64 VGPRs)
    int grow = row0 + wv * 16 + m;
    if (grow >= N) grow = N - 1;     // clamp reads; stores guarded below
    const float2* arow = (const float2*)(A + (size_t)grow * D);
    float2 af[32];
#pragma unroll
    for (int t = 0; t < 32; ++t) af[t] = arow[t * 2 + hi];   // K = 4t + 2*hi

    v8f acc[8];
    {
        v8f z = {};
#pragma unroll
        for (int nt = 0; nt < 8; ++nt) acc[nt] = z;
    }

    for (int t = 0; t < 32; ++t) {               // K loop, 4 per step
        v2f a;
        a.x = af[t].x;
        a.y = af[t].y;
        // k-pair index for this half-wave: kb = 4t + 2*hi -> pair = 2t + hi
        const float2* wp = (const float2*)&ldsW[(t * 2 + hi) * 256];
#pragma unroll
        for (int nt = 0; nt < 8; ++nt) {         // 8 independent 16x16 tiles
            const float2 bv = wp[nt * 16 + m];   // one ds_load_b64
            v2f b;
            b.x = bv.x;
            b.y = bv.y;
            acc[nt] = __builtin_amdgcn_wmma_f32_16x16x4_f32(
                false, a, false, b, (short)0, acc[nt], false, false);
        }
    }

    // epilogue: bias + SELU + guarded store
    // C/D layout: VGPR j -> row (j + 8*hi), lane&15 -> col within tile
    const int rbase = row0 + wv * 16 + hi * 8;
#pragma unroll
    for (int nt = 0; nt < 8; ++nt) {
        const int col = nt * 16 + m;
        const float bsc = bias[col];
#pragma unroll
        for (int j = 0; j < 8; ++j) {
            const int gr = rbase + j;
            if (gr < N)
                out[(size_t)gr * out_row_stride + col] = selu_f(acc[nt][j] + bsc);
        }
    }
}

extern "C" void kernel_launch(void* const* d_in, const int* in_sizes, int n_in,
                              void* d_out, int out_size, void* d_ws, size_t ws_size,
                              hipStream_t stream) {
    const float* x_in = (const float*)d_in[0];
    const int*   erow = (const int*)d_in[1];
    const int*   ecol = (const int*)d_in[2];
    const float* eval = (const float*)d_in[3];
    const float* W1   = (const float*)d_in[4];
    const float* b1   = (const float*)d_in[5];
    const float* W2   = (const float*)d_in[6];
    const float* b2   = (const float*)d_in[7];

    const int N = in_sizes[0] / D;        // 100000
    const int E = in_sizes[1];            // 1600000
    float* out = (float*)d_out;

    // workspace layout: h0[2][N][D] | h1[2][N][D] | row_ptr[N+1]
    const size_t branch_elems = (size_t)N * D;
    float* h0 = (float*)d_ws;
    float* h1 = h0 + 2 * branch_elems;
    int* row_ptr = (int*)(h1 + 2 * branch_elems);

    // 1) CSR offsets
    rowptr_kernel<<<(N + 1 + 255) / 256, 256, 0, stream>>>(erow, row_ptr, N, E);

    // 2) SPMM + residual for both branches
    spmm_kernel<<<(N + 7) / 8, 256, 0, stream>>>(x_in, row_ptr, ecol, eval, h0, N);

    // 3) two fused GEMM+SELU layers, both branches via grid.y
    dim3 g((N + 127) / 128, 2);
    // layer 1: h1 = selu(h0 @ W1 + b1)   (per-branch dense [N,128])
    gemm_selu_kernel<<<g, 256, 0, stream>>>(h0, branch_elems, W1, b1,
                                            h1, D, branch_elems, N);
    // layer 2: out[:, br*128:(br+1)*128] = selu(h1 @ W2 + b2)
    gemm_selu_kernel<<<g, 256, 0, stream>>>(h1, branch_elems, W2, b2,
                                            out, 2 * D, (size_t)D, N);
}